// MatrixGraphConvolution_46162308497840
// MI455X (gfx1250) — compile-verified
//
#include <hip/hip_runtime.h>

typedef float v2f __attribute__((ext_vector_type(2)));
typedef float v8f __attribute__((ext_vector_type(8)));

#define N_NODES   50000
#define DEGREE    16
#define D_FEAT    96
#define TILE_M    16
#define LDS_STRIDE 98   // even (keeps float2 8B alignment); 16 rows hit distinct bank pairs mod 64

__global__ __launch_bounds__(192)
void MatrixGraphConvolution_kernel(const float* __restrict__ x,
                                   const int*   __restrict__ edge_src,   // first E entries of edge_index
                                   const float* __restrict__ W,
                                   const float* __restrict__ Bm,
                                   float*       __restrict__ out)
{
    __shared__ float hTile[TILE_M * LDS_STRIDE];   // mean-aggregated neighbor features
    __shared__ float xTile[TILE_M * LDS_STRIDE];   // self features
    __shared__ int   srcIdx[TILE_M * DEGREE];

    const int tid   = threadIdx.x;
    const int node0 = blockIdx.x * TILE_M;

    // Stage the 256 source indices (edges are grouped by destination: node n -> edges [16n,16n+16))
    for (int i = tid; i < TILE_M * DEGREE; i += 192)
        srcIdx[i] = edge_src[node0 * DEGREE + i];
    __syncthreads();

    // Aggregate: h[n,f] = (1/16) * sum_k x[src[n,k], f]; gathers hit L2 (x is 19.2MB << 192MB)
    const float rdeg = 1.0f / (float)DEGREE;
    for (int i = tid; i < TILE_M * D_FEAT; i += 192) {
        const int n = i / D_FEAT;
        const int f = i - n * D_FEAT;
        float s = 0.0f;
        #pragma unroll
        for (int k = 0; k < DEGREE; ++k)
            s += x[(size_t)srcIdx[n * DEGREE + k] * D_FEAT + f];
        hTile[n * LDS_STRIDE + f] = s * rdeg;
        xTile[n * LDS_STRIDE + f] = x[(size_t)(node0 + n) * D_FEAT + f];
    }
    __syncthreads();

    // 6 waves, wave w computes out[node0:node0+16, 16w:16w+16] in full fp32 via V_WMMA_F32_16X16X4_F32
    const int wave = tid >> 5;             // 0..5
    const int lane = tid & 31;
    const int mrow = lane & 15;            // row within half-wave
    const int kOff = (lane < 16) ? 0 : 2;  // f32 A/B frag: lanes 0-15 hold K,K+1; lanes 16-31 hold K+2,K+3
    const int oCol = wave * 16 + mrow;     // output feature this lane supplies B-frag data for

    v8f acc = {};
    // neighbor term: h @ W^T  (B-frag[k][o] = W[o][k], contiguous pair in row-major W)
    for (int k0 = 0; k0 < D_FEAT; k0 += 4) {
        v2f a = *(const v2f*)&hTile[mrow * LDS_STRIDE + k0 + kOff];
        v2f b = *(const v2f*)&W[(size_t)oCol * D_FEAT + k0 + kOff];
        acc = __builtin_amdgcn_wmma_f32_16x16x4_f32(false, a, false, b, (short)0, acc, false, false);
    }
    // self term: x @ B^T
    for (int k0 = 0; k0 < D_FEAT; k0 += 4) {
        v2f a = *(const v2f*)&xTile[mrow * LDS_STRIDE + k0 + kOff];
        v2f b = *(const v2f*)&Bm[(size_t)oCol * D_FEAT + k0 + kOff];
        acc = __builtin_amdgcn_wmma_f32_16x16x4_f32(false, a, false, b, (short)0, acc, false, false);
    }

    // C/D 16x16 f32 layout: VGPR r = row M=r (lanes 0-15) / M=r+8 (lanes 16-31), N = lane&15
    const int rowHi = (lane < 16) ? 0 : 8;
    #pragma unroll
    for (int r = 0; r < 8; ++r)
        out[(size_t)(node0 + r + rowHi) * D_FEAT + wave * 16 + mrow] = acc[r];
}

extern "C" void kernel_launch(void* const* d_in, const int* in_sizes, int n_in,
                              void* d_out, int out_size, void* d_ws, size_t ws_size,
                              hipStream_t stream) {
    const float* x    = (const float*)d_in[0];   // [N, 96]
    const int*   edge = (const int*)d_in[1];     // [2, E]; row 0 = src
    const float* W    = (const float*)d_in[2];   // [96, 96]
    const float* Bm   = (const float*)d_in[3];   // [96, 96]
    float*       out  = (float*)d_out;           // [N, 96]

    const int nBlocks = (N_NODES + TILE_M - 1) / TILE_M;   // 3125
    MatrixGraphConvolution_kernel<<<nBlocks, 192, 0, stream>>>(x, edge, W, Bm, out);
}